// RNN_21543555957078
// MI455X (gfx1250) — compile-verified
//
#include <hip/hip_runtime.h>
#include <hip/hip_bf16.h>

typedef __attribute__((ext_vector_type(2))) float v2f;
typedef __attribute__((ext_vector_type(8))) float v8f;

#define BATCH 8192
#define SEQ   2048

// Branchless lane n <-> lane n^16 swap: v_permlanex16_b32 with identity selects.
// Pure VALU (no LDS, no DScnt wait, no exec-mask branching).
__device__ __forceinline__ float halfswap(float v) {
    int i = __float_as_int(v);
    int r = __builtin_amdgcn_permlanex16(i, i, 0x76543210, 0xFEDCBA98, false, false);
    return __int_as_float(r);
}

// One wave = 4 "streams" of 16 chains each (64 chains per wave, 128 waves total).
// Per stream per layer per step: one V_WMMA_F32_16X16X4_F32 computing
//   Z^T = [Wh^T | Uh^T] (16x4, rows>=2 zero)  @  [inp^T ; h^T] (4x16)
// A-operand layout (16x4 f32): lanes0-15 -> K=0,1 in v0,v1 ; lanes16-31 -> K=2,3
// B-operand layout (4x16 f32): lanes0-15 -> rows K=0,1 ; lanes16-31 -> rows K=2,3
// D layout: VGPR r, lanes0-15 = row M=r  => z0 in d.s0, z1 in d.s1, lane = chain.
// 4 independent chains-groups per wave give ~4-way ILP so the SIMD can issue
// other streams' VALU/WMMA during each stream's wmma->exp->rcp->wmma dep chain.
__global__ __launch_bounds__(256) void rnn_wmma_kernel(
    const float* __restrict__ x, const float* __restrict__ means,
    const float* __restrict__ log_vars, const float* __restrict__ Wh,
    const float* __restrict__ Uh, float* __restrict__ out)
{
    const int lane  = threadIdx.x & 31;
    const int wid   = (blockIdx.x * blockDim.x + threadIdx.x) >> 5;
    const int base  = wid * 64;               // 64 batch chains per wave
    const int m     = lane & 15;              // matrix row within half-wave
    const bool upper = (lane >> 4) != 0;      // false -> K rows 0..1, true -> K rows 2..3

    // Gaussian embedding constants (var = exp(log_var))
    const float mean0 = means[0], mean1 = means[1];
    const float lv0 = log_vars[0], lv1 = log_vars[1];
    const float iv0 = __expf(-lv0), iv1 = __expf(-lv1);
    const float HALF_LOG_2PI = 0.91893853320467274178f;
    const float n0 = HALF_LOG_2PI + 0.5f * lv0;
    const float n1 = HALF_LOG_2PI + 0.5f * lv1;

    // Constant A operands: A_l = [Wh_l^T | Uh_l^T] (only rows m<2 nonzero)
    v2f A[2];
#pragma unroll
    for (int l = 0; l < 2; ++l) {
        float w0 = 0.0f, w1 = 0.0f;
        if (m < 2) {
            const float* Wsel = upper ? (Uh + l * 4) : (Wh + l * 4);
            w0 = Wsel[0 * 2 + m];   // K = {0 or 2} row, column m
            w1 = Wsel[1 * 2 + m];   // K = {1 or 3} row, column m
        }
        A[l].x = w0;
        A[l].y = w1;
    }

    // Persistent hidden state per stream, stored UPPER-RESIDENT (already
    // half-swapped so it drops straight into the B operand's K=2,3 rows).
    float h0l0s[4] = {0, 0, 0, 0}, h1l0s[4] = {0, 0, 0, 0};
    float h0l1s[4] = {0, 0, 0, 0}, h1l1s[4] = {0, 0, 0, 0};

    const float* xrowA = x + (size_t)(base + lane) * SEQ;        // streams 0,1
    const float* xrowB = x + (size_t)(base + 32 + lane) * SEQ;   // streams 2,3
    float* orowA = out + (size_t)(base + lane) * SEQ * 2;
    float* orowB = out + (size_t)(base + 32 + lane) * SEQ * 2;

    v8f cz = {};      // zero C accumulator
    float4 obufA, obufB;

    // Double-buffered x stream: issue next group's load before crunching the
    // current 4 steps so s_wait_loadcnt lands after ~16 WMMA dep-chains.
    float4 curA = *reinterpret_cast<const float4*>(xrowA);
    float4 curB = *reinterpret_cast<const float4*>(xrowB);

    for (int t = 0; t < SEQ; t += 4) {
        const int tn = (t + 4 < SEQ) ? (t + 4) : t;   // clamped (uniform scalar select)
        const float4 nxtA = *reinterpret_cast<const float4*>(xrowA + tn);
        const float4 nxtB = *reinterpret_cast<const float4*>(xrowB + tn);
        __builtin_prefetch(xrowA + t + 64, 0, 3);     // pull into near caches
        __builtin_prefetch(xrowB + t + 64, 0, 3);
#pragma unroll
        for (int k = 0; k < 4; ++k) {
            const float xa = (k == 0) ? curA.x : (k == 1) ? curA.y : (k == 2) ? curA.z : curA.w;
            const float xb = (k == 0) ? curB.x : (k == 1) ? curB.y : (k == 2) ? curB.z : curB.w;
            // odd streams' x lives in lanes16-31; bring it to lanes0-15
            const float xs[4] = { xa, halfswap(xa), xb, halfswap(xb) };
            float q0p[4], q1p[4];   // layer-1 outputs, plain (lanes0-15 valid)
#pragma unroll
            for (int s = 0; s < 4; ++s) {
                // Gaussian neg-log-prob embedding (lanes 0-15 meaningful)
                const float dx0 = xs[s] - mean0, dx1 = xs[s] - mean1;
                const float e0 = n0 + dx0 * dx0 * iv0;
                const float e1 = n1 + dx1 * dx1 * iv1;

                // ---- layer 0: B = [emb^T ; h_l0^T] ----
                v2f b0;
                b0.x = upper ? h0l0s[s] : e0;
                b0.y = upper ? h1l0s[s] : e1;
                v8f d0 = __builtin_amdgcn_wmma_f32_16x16x4_f32(
                    false, A[0], false, b0, (short)0, cz, false, false);
                // 2-way softmax: h0 = 1/(1+e^{z1-z0}), h1 = e*h0
                float g = __expf(d0.s1 - d0.s0);
                float r = __builtin_amdgcn_rcpf(1.0f + g);
                const float p0 = r, p1 = g * r;

                // ---- layer 1: B = [h_l0(t)^T ; h_l1(t-1)^T] ----
                v2f b1;
                b1.x = upper ? h0l1s[s] : p0;
                b1.y = upper ? h1l1s[s] : p1;
                // refresh layer-0 recurrent state (swap once, reuse next step)
                h0l0s[s] = halfswap(p0);
                h1l0s[s] = halfswap(p1);
                v8f d1 = __builtin_amdgcn_wmma_f32_16x16x4_f32(
                    false, A[1], false, b1, (short)0, cz, false, false);
                g = __expf(d1.s1 - d1.s0);
                r = __builtin_amdgcn_rcpf(1.0f + g);
                const float q0 = r, q1 = g * r;
                q0p[s] = q0; q1p[s] = q1;
                // swapped copies serve BOTH next-step B1 upper half AND output merge
                h0l1s[s] = halfswap(q0);
                h1l1s[s] = halfswap(q1);
            }
            // Merge stream pairs: lanes0-15 = even stream (plain),
            // lanes16-31 = odd stream (already-swapped state)
            const float oA0 = upper ? h0l1s[1] : q0p[0];
            const float oA1 = upper ? h1l1s[1] : q1p[0];
            const float oB0 = upper ? h0l1s[3] : q0p[2];
            const float oB1 = upper ? h1l1s[3] : q1p[2];
            if (k & 1) {
                obufA.z = oA0; obufA.w = oA1;   // steps (t+k-1, t+k) -> one 16B store
                obufB.z = oB0; obufB.w = oB1;
                *reinterpret_cast<float4*>(orowA + (size_t)(t + k - 1) * 2) = obufA;
                *reinterpret_cast<float4*>(orowB + (size_t)(t + k - 1) * 2) = obufB;
            } else {
                obufA.x = oA0; obufA.y = oA1;
                obufB.x = oB0; obufB.y = oB1;
            }
        }
        curA = nxtA;
        curB = nxtB;
    }
}

extern "C" void kernel_launch(void* const* d_in, const int* in_sizes, int n_in,
                              void* d_out, int out_size, void* d_ws, size_t ws_size,
                              hipStream_t stream) {
    const float* x        = (const float*)d_in[0];
    const float* means    = (const float*)d_in[1];
    const float* log_vars = (const float*)d_in[2];
    const float* Wh       = (const float*)d_in[3];
    const float* Uh       = (const float*)d_in[4];
    float* out = (float*)d_out;
    // 64 chains/wave * 8 waves/block = 512 chains per block -> 16 blocks
    rnn_wmma_kernel<<<dim3(BATCH / 512), dim3(256), 0, stream>>>(
        x, means, log_vars, Wh, Uh, out);
}